// CoordCrossAtt_30537217475311
// MI455X (gfx1250) — compile-verified
//
#include <hip/hip_runtime.h>
#include <hip/hip_bf16.h>
#include <math.h>

typedef __attribute__((ext_vector_type(2))) float v2f;
typedef __attribute__((ext_vector_type(8))) float v8f;

#define N_B   32
#define INP_C 512
#define MIP_C 16
#define OUP_C 512
#define HH    64
#define WW    64

static __device__ __forceinline__ v8f wmma_f32(v2f a, v2f b, v8f c) {
    // D = A(16x4, f32) * B(4x16, f32) + C(16x16, f32)
    return __builtin_amdgcn_wmma_f32_16x16x4_f32(false, a, false, b, (short)0, c, false, false);
}

// ---------------------------------------------------------------------------
// Kernel 1: fused H-mean and W-mean pooling.  One block per (n, c) plane.
// Reads x once (16 KiB plane into LDS), writes y_t[n][l][c], l in [0,128):
//   l in [0,64)  : x_h  (mean over W)  -> feeds q path
//   l in [64,128): x_w  (mean over H)  -> feeds k/v path
// Transposed (c innermost) so kernel 2's WMMA B fragments are contiguous.
// ---------------------------------------------------------------------------
__global__ __launch_bounds__(256) void pool_kernel(const float* __restrict__ x,
                                                   float* __restrict__ y_t) {
    __shared__ float plane[HH * WW];
    const int c = blockIdx.x, n = blockIdx.y, tid = threadIdx.x;
    const float4* src = (const float4*)(x + (((size_t)n * INP_C + c) << 12));
    float4* pl4 = (float4*)plane;
#pragma unroll
    for (int i = 0; i < 4; ++i) pl4[i * 256 + tid] = src[i * 256 + tid];
    __syncthreads();

    float* dst = y_t + (size_t)n * 128 * INP_C + c;
    if (tid < 64) {                       // row mean (over w), rotated to dodge bank conflicts
        const int h = tid;
        float s = 0.f;
#pragma unroll 8
        for (int w = 0; w < 64; ++w) s += plane[h * 64 + ((w + h) & 63)];
        dst[(size_t)h * INP_C] = s * (1.0f / 64.0f);
    } else if (tid < 128) {               // column mean (over h), conflict-free
        const int w = tid - 64;
        float s = 0.f;
#pragma unroll 8
        for (int h = 0; h < 64; ++h) s += plane[h * 64 + w];
        dst[(size_t)(64 + w) * INP_C] = s * (1.0f / 64.0f);
    }
}

// ---------------------------------------------------------------------------
// Kernel 2: one block (8 waves) per batch n.  All-WMMA GEMM chain in LDS:
//   cv1(16x512x128) -> q/k/v(16x16x64) -> attn(64x16x64)+softmax -> z(16x64x64)
// Stores z transposed: z_t[n][h][d] (d contiguous for kernel 3's B fragments).
// ---------------------------------------------------------------------------
__global__ __launch_bounds__(256) void attn_core(
    const float* __restrict__ y_t,
    const float* __restrict__ w_cv1, const float* __restrict__ b_cv1,
    const float* __restrict__ w_q,   const float* __restrict__ b_q,
    const float* __restrict__ w_k,   const float* __restrict__ b_k,
    const float* __restrict__ w_v,   const float* __restrict__ b_v,
    float* __restrict__ z_t) {
    __shared__ float cv1_s[MIP_C][128];
    __shared__ float q_s[MIP_C][64];
    __shared__ float k_s[MIP_C][64];
    __shared__ float v_s[MIP_C][64];
    __shared__ float attn_s[64][64];

    const int n    = blockIdx.x;
    const int tid  = threadIdx.x;
    const int wave = tid >> 5;
    const int lane = tid & 31;
    const int lm    = lane & 15;        // M (or N) index within 16
    const int khalf = (lane >> 4) & 1;  // K-half for A/B frags, M-half for C/D
    const int kb    = khalf * 2;

    // ---- Stage A: cv1  D(16x128) = w_cv1(16x512) * y(512x128) + b ----------
    {
        const int col = wave * 16 + lm;                       // l in [0,128)
        v8f acc;
#pragma unroll
        for (int r = 0; r < 8; ++r) acc[r] = b_cv1[r + 8 * khalf];
        const float* arow = w_cv1 + lm * INP_C;               // A[m=lm][k]
        const float* bcol = y_t + ((size_t)n * 128 + col) * INP_C;  // B[k][col]
        for (int k4 = 0; k4 < 128; ++k4) {
            const int kk = k4 * 4 + kb;
            v2f a; a[0] = arow[kk]; a[1] = arow[kk + 1];
            v2f b; b[0] = bcol[kk]; b[1] = bcol[kk + 1];
            acc = wmma_f32(a, b, acc);
        }
#pragma unroll
        for (int r = 0; r < 8; ++r) cv1_s[r + 8 * khalf][col] = acc[r];
    }
    __syncthreads();

    // ---- Stage B: q/k/v  (16x16x64 each) -> 12 tiles over 8 waves ----------
    for (int t = wave; t < 12; t += 8) {
        const int which = t >> 2;                 // 0=q, 1=k, 2=v
        const int ct    = t & 3;                  // 16-wide position tile
        const float* wm = (which == 0) ? w_q : (which == 1) ? w_k : w_v;
        const float* bm = (which == 0) ? b_q : (which == 1) ? b_k : b_v;
        const int loff  = (which == 0) ? 0 : 64;  // q uses yh, k/v use yw
        const int col   = ct * 16 + lm;
        v8f acc;
#pragma unroll
        for (int r = 0; r < 8; ++r) acc[r] = bm[r + 8 * khalf];
        const float* arow = wm + lm * MIP_C;
#pragma unroll
        for (int k4 = 0; k4 < 4; ++k4) {
            const int kk = k4 * 4 + kb;
            v2f a; a[0] = arow[kk];               a[1] = arow[kk + 1];
            v2f b; b[0] = cv1_s[kk][loff + col];  b[1] = cv1_s[kk + 1][loff + col];
            acc = wmma_f32(a, b, acc);
        }
        float (*dst)[64] = (which == 0) ? q_s : (which == 1) ? k_s : v_s;
#pragma unroll
        for (int r = 0; r < 8; ++r) dst[r + 8 * khalf][col] = acc[r];
    }
    __syncthreads();

    // ---- Stage C: attn(64x64) = q^T k * 0.25  -> 16 tiles over 8 waves -----
    for (int t = wave; t < 16; t += 8) {
        const int mt = t >> 2, nt = t & 3;
        const int hA = mt * 16 + lm;              // A row: h
        const int wB = nt * 16 + lm;              // B col: w
        v8f acc = {};
#pragma unroll
        for (int k4 = 0; k4 < 4; ++k4) {
            const int kk = k4 * 4 + kb;           // d index
            v2f a; a[0] = q_s[kk][hA]; a[1] = q_s[kk + 1][hA];
            v2f b; b[0] = k_s[kk][wB]; b[1] = k_s[kk + 1][wB];
            acc = wmma_f32(a, b, acc);
        }
#pragma unroll
        for (int r = 0; r < 8; ++r)
            attn_s[mt * 16 + r + 8 * khalf][nt * 16 + lm] = acc[r] * 0.25f;
    }
    __syncthreads();

    // ---- Stage D: softmax over w, one thread per row h ---------------------
    if (tid < 64) {
        float mx = -INFINITY;
        for (int w = 0; w < 64; ++w) mx = fmaxf(mx, attn_s[tid][w]);
        float s = 0.f;
        for (int w = 0; w < 64; ++w) { float e = __expf(attn_s[tid][w] - mx); attn_s[tid][w] = e; s += e; }
        const float inv = 1.0f / s;
        for (int w = 0; w < 64; ++w) attn_s[tid][w] *= inv;
    }
    __syncthreads();

    // ---- Stage E: z(16x64) = v(16x64) @ attn^T(64x64) -> 4 tiles, waves 0-3 -
    if (wave < 4) {
        const int hcol = wave * 16 + lm;          // N index: h
        v8f acc = {};
#pragma unroll 4
        for (int k4 = 0; k4 < 16; ++k4) {
            const int kk = k4 * 4 + kb;           // w index
            v2f a; a[0] = v_s[lm][kk];      a[1] = v_s[lm][kk + 1];
            v2f b; b[0] = attn_s[hcol][kk]; b[1] = attn_s[hcol][kk + 1];
            acc = wmma_f32(a, b, acc);
        }
#pragma unroll
        for (int r = 0; r < 8; ++r)
            z_t[((size_t)n * 64 + hcol) * MIP_C + r + 8 * khalf] = acc[r];
    }
}

// ---------------------------------------------------------------------------
// Kernel 3: y_att = sigmoid(w_proj @ z + b) via WMMA (wave 0-3), then the
// bandwidth-dominant fused elementwise: out = x * y_att, float4 streaming.
// One block per (ctile of 16 channels, n): 256 KiB read + 256 KiB write.
// ---------------------------------------------------------------------------
__global__ __launch_bounds__(256) void proj_mul(
    const float* __restrict__ x, const float* __restrict__ z_t,
    const float* __restrict__ w_proj, const float* __restrict__ b_proj,
    float* __restrict__ out) {
    __shared__ float yatt_s[16][64];
    const int n = blockIdx.y, c0 = blockIdx.x * 16;
    const int tid = threadIdx.x, wave = tid >> 5, lane = tid & 31;
    const int lm = lane & 15, khalf = (lane >> 4) & 1, kb = khalf * 2;

    if (wave < 4) {                                // M=16(c) x K=16(m) x N=64(h)
        const int hcol = wave * 16 + lm;
        v8f acc;
#pragma unroll
        for (int r = 0; r < 8; ++r) acc[r] = b_proj[c0 + r + 8 * khalf];
        const float* arow = w_proj + (size_t)(c0 + lm) * MIP_C;
        const float* brow = z_t + ((size_t)n * 64 + hcol) * MIP_C;
#pragma unroll
        for (int k4 = 0; k4 < 4; ++k4) {
            const int kk = k4 * 4 + kb;
            v2f a; a[0] = arow[kk]; a[1] = arow[kk + 1];
            v2f b; b[0] = brow[kk]; b[1] = brow[kk + 1];
            acc = wmma_f32(a, b, acc);
        }
#pragma unroll
        for (int r = 0; r < 8; ++r)
            yatt_s[r + 8 * khalf][hcol] = 1.0f / (1.0f + __expf(-acc[r]));
    }
    __syncthreads();

    // 16 channels x 64x64 plane = 65536 floats; a float4 spans 4 w's (same c,h)
    const size_t base = ((size_t)n * OUP_C + c0) << 12;
    const float4* xin = (const float4*)(x + base);
    float4* xout = (float4*)(out + base);
#pragma unroll 4
    for (int i = 0; i < 64; ++i) {
        const int e4 = i * 256 + tid;
        const int e  = e4 << 2;
        const float s = yatt_s[e >> 12][(e >> 6) & 63];
        float4 xv = xin[e4];
        xv.x *= s; xv.y *= s; xv.z *= s; xv.w *= s;
        xout[e4] = xv;
    }
}

extern "C" void kernel_launch(void* const* d_in, const int* in_sizes, int n_in,
                              void* d_out, int out_size, void* d_ws, size_t ws_size,
                              hipStream_t stream) {
    const float* x      = (const float*)d_in[0];
    const float* w_cv1  = (const float*)d_in[1];
    const float* b_cv1  = (const float*)d_in[2];
    const float* w_q    = (const float*)d_in[3];
    const float* b_q    = (const float*)d_in[4];
    const float* w_k    = (const float*)d_in[5];
    const float* b_k    = (const float*)d_in[6];
    const float* w_v    = (const float*)d_in[7];
    const float* b_v    = (const float*)d_in[8];
    const float* w_proj = (const float*)d_in[9];
    const float* b_proj = (const float*)d_in[10];
    float* out = (float*)d_out;

    float* y_t = (float*)d_ws;                         // [32][128][512] = 8 MiB
    float* z_t = y_t + (size_t)N_B * 128 * INP_C;      // [32][64][16]   = 128 KiB

    pool_kernel<<<dim3(INP_C, N_B), 256, 0, stream>>>(x, y_t);
    attn_core<<<N_B, 256, 0, stream>>>(y_t, w_cv1, b_cv1, w_q, b_q, w_k, b_k,
                                       w_v, b_v, z_t);
    proj_mul<<<dim3(OUP_C / 16, N_B), 256, 0, stream>>>(x, z_t, w_proj, b_proj, out);
}